// SparseGatingNetwork_77730318123206
// MI455X (gfx1250) — compile-verified
//
#include <hip/hip_runtime.h>
#include <math.h>

typedef __attribute__((ext_vector_type(2))) float v2f;
typedef __attribute__((ext_vector_type(8))) float v8f;

namespace {
constexpr int LDIM = 12;            // router input features
constexpr int DDIM = 64;            // hidden dim
constexpr int EDIM = 8;             // experts
constexpr int TPT  = 16;            // tokens per tile (WMMA M)
constexpr int WAVES = 8;            // waves per block
constexpr int HSTRIDE = 68;         // padded LDS stride for h tile (bank-conflict-free)
}

__global__ __launch_bounds__(WAVES * 32)
void moe_router_wmma_f32(const float* __restrict__ x,   // [B*N, 12]
                         const float* __restrict__ W1,  // [12, 64]
                         const float* __restrict__ b1,  // [64]
                         const float* __restrict__ W2,  // [64, 8]
                         const float* __restrict__ b2,  // [8]
                         float* __restrict__ out,       // [B*N, 8]
                         int numTiles)
{
    __shared__ float w2p[DDIM * 16];                  // W2 zero-padded to 64x16
    __shared__ float b2p[16];                         // b2 zero-padded to 16
    __shared__ float xs_all[WAVES][TPT * LDIM];       // staged x tiles
    __shared__ float hs_all[WAVES][TPT * HSTRIDE];    // h tiles (layout shuffle)
    __shared__ float ls_all[WAVES][TPT * 16];         // logits (padded cols)

    // Block-cooperative: zero-pad W2/b2 into LDS (keeps GEMM2 branch-free).
    for (int i = threadIdx.x; i < DDIM * 16; i += blockDim.x) {
        int k = i >> 4, c = i & 15;
        w2p[i] = (c < EDIM) ? W2[k * EDIM + c] : 0.0f;
    }
    if (threadIdx.x < 16)
        b2p[threadIdx.x] = (threadIdx.x < EDIM) ? b2[threadIdx.x] : 0.0f;
    __syncthreads();

    const int lane = threadIdx.x & 31;
    const int wave = threadIdx.x >> 5;
    const int tile = blockIdx.x * WAVES + wave;   // uniform per wave
    if (tile >= numTiles) return;                 // after the only barrier

    const int tokenBase = tile * TPT;
    float* xs = xs_all[wave];
    float* hs = hs_all[wave];
    float* ls = ls_all[wave];

    // ---- Stage x tile (16x12 f32 = 192 floats) fully coalesced: 6 floats/lane.
    {
        const float* src = x + (size_t)tokenBase * LDIM;
        const int o = lane * 6;                   // 24B/lane, 8B aligned
        v2f a0 = *(const v2f*)(src + o);
        v2f a1 = *(const v2f*)(src + o + 2);
        v2f a2 = *(const v2f*)(src + o + 4);
        *(v2f*)(xs + o)     = a0;
        *(v2f*)(xs + o + 2) = a1;
        *(v2f*)(xs + o + 4) = a2;
    }

    // WMMA f32 16x16x4 operand indexing (ISA 7.12.2):
    //   A: lanes 0-15 -> M=lane, VGPR0=K0,VGPR1=K1; lanes 16-31 -> M=lane-16, K2/K3.
    //   B: lanes hold column N=lane&15; half-wave selects K pair (mirrors A).
    const int m  = lane & 15;          // A row (token) / B,C,D column
    const int kb = (lane >> 4) * 2;    // K sub-offset for this half-wave

    // ---- GEMM1: h[16x64] = relu(x[16x12] @ W1[12x64] + b1), 12 f32 WMMAs.
    v8f acc[4] = {};
    #pragma unroll
    for (int ki = 0; ki < 3; ++ki) {
        const int k0 = ki * 4 + kb;
        v2f a; a.x = xs[m * LDIM + k0]; a.y = xs[m * LDIM + k0 + 1];
        #pragma unroll
        for (int nt = 0; nt < 4; ++nt) {
            const int d = nt * 16 + m;
            v2f b; b.x = W1[k0 * DDIM + d]; b.y = W1[(k0 + 1) * DDIM + d];
            acc[nt] = __builtin_amdgcn_wmma_f32_16x16x4_f32(
                false, a, false, b, (short)0, acc[nt], false, false);
        }
    }

    // ---- Bias + ReLU, then C/D-layout -> A-layout via LDS (padded stride).
    #pragma unroll
    for (int nt = 0; nt < 4; ++nt) {
        const float bv = b1[nt * 16 + m];
        #pragma unroll
        for (int r = 0; r < 8; ++r) {
            const int row = r + ((lane >> 4) << 3);   // lanes 16-31 hold M=8+r
            float v = acc[nt][r] + bv;
            hs[row * HSTRIDE + nt * 16 + m] = (v > 0.0f) ? v : 0.0f;
        }
    }

    // ---- GEMM2: logits[16x16(pad)] = h[16x64] @ W2pad[64x16], 16 f32 WMMAs.
    v8f acc2 = {};
    #pragma unroll
    for (int ki = 0; ki < 16; ++ki) {
        const int k0 = ki * 4 + kb;
        v2f a; a.x = hs[m * HSTRIDE + k0]; a.y = hs[m * HSTRIDE + k0 + 1];
        v2f b; b.x = w2p[k0 * 16 + m];     b.y = w2p[(k0 + 1) * 16 + m];
        acc2 = __builtin_amdgcn_wmma_f32_16x16x4_f32(
            false, a, false, b, (short)0, acc2, false, false);
    }

    // ---- Logits (+b2) to LDS (padded 16 cols; no divergence needed).
    {
        const float bv = b2p[m];
        #pragma unroll
        for (int r = 0; r < 8; ++r) {
            const int row = r + ((lane >> 4) << 3);
            ls[row * 16 + m] = acc2[r] + bv;
        }
    }

    // ---- Top-2 + softmax + scatter: lanes 0-15 handle one token each.
    if (lane < 16) {
        float lg[EDIM];
        #pragma unroll
        for (int e = 0; e < EDIM; ++e) lg[e] = ls[lane * 16 + e];

        int i1 = 0; float v1 = lg[0];
        #pragma unroll
        for (int e = 1; e < EDIM; ++e)            // strict > keeps lowest index on ties
            if (lg[e] > v1) { v1 = lg[e]; i1 = e; }
        int i2 = -1; float v2 = -INFINITY;
        #pragma unroll
        for (int e = 0; e < EDIM; ++e)
            if (e != i1 && lg[e] > v2) { v2 = lg[e]; i2 = e; }

        const float e2 = __expf(v2 - v1);         // v1 >= v2, stable softmax
        const float inv = 1.0f / (1.0f + e2);
        const float g1 = inv, g2 = e2 * inv;

        float w[EDIM];
        #pragma unroll
        for (int e = 0; e < EDIM; ++e)
            w[e] = (e == i1) ? g1 : ((e == i2) ? g2 : 0.0f);

        float4* dst = (float4*)(out + (size_t)(tokenBase + lane) * EDIM);
        dst[0] = make_float4(w[0], w[1], w[2], w[3]);
        dst[1] = make_float4(w[4], w[5], w[6], w[7]);
    }
}

extern "C" void kernel_launch(void* const* d_in, const int* in_sizes, int n_in,
                              void* d_out, int out_size, void* d_ws, size_t ws_size,
                              hipStream_t stream) {
    const float* x  = (const float*)d_in[0];
    const float* W1 = (const float*)d_in[1];
    const float* b1 = (const float*)d_in[2];
    const float* W2 = (const float*)d_in[3];
    const float* b2 = (const float*)d_in[4];
    float* out = (float*)d_out;

    const int numTokens = in_sizes[0] / LDIM;       // B*N = 524288
    const int numTiles  = numTokens / TPT;          // 32768 (exact)
    const int blocks    = (numTiles + WAVES - 1) / WAVES;

    moe_router_wmma_f32<<<blocks, WAVES * 32, 0, stream>>>(
        x, W1, b1, W2, b2, out, numTiles);
}